// SwarmNet_20641612824891
// MI455X (gfx1250) — compile-verified
//
#include <hip/hip_runtime.h>
#include <math.h>

// SwarmNet on MI455X (gfx1250).
// Dominant work: Y_l = A_l(8192x8192, fp32) @ Xcat(8192x16) -> maps to
// V_WMMA_F32_16X16X4_F32 with a single 16-wide N tile. HBM-bound:
// 4 steps x 3 x 256MB = 3.2 GB of A traffic -> ~138us floor at 23.3 TB/s.

typedef __attribute__((ext_vector_type(2))) float v2f;
typedef __attribute__((ext_vector_type(8))) float v8f;

#define N_NODES 8192
#define BATCH   4
#define FDIM    4
#define NCOLS   16   // BATCH * FDIM

// ---------------------------------------------------------------------------
// Pack X0 = time_segs[:,0] into Xcat (N x 16 row-major, col = b*F+f) and
// XbT (16 x N), the transposed copy used as the WMMA B operand.
// ---------------------------------------------------------------------------
__global__ void pack_x0(const float* __restrict__ ts,
                        float* __restrict__ xcat,
                        float* __restrict__ xbt) {
    int n = blockIdx.x * blockDim.x + threadIdx.x;
    if (n >= N_NODES) return;
#pragma unroll
    for (int b = 0; b < BATCH; ++b) {
        const float* src = ts + ((size_t)b * N_NODES + n) * FDIM; // [b][0][n][f]
#pragma unroll
        for (int f = 0; f < FDIM; ++f) {
            float v = src[f];
            int c = b * FDIM + f;
            xcat[(size_t)n * NCOLS + c] = v;
            xbt[(size_t)c * N_NODES + n] = v;
        }
    }
}

// ---------------------------------------------------------------------------
// Y_l = A_l @ Xcat  for l = 0,1,2.  One wave per (layer, 16-row tile).
// A fragment (16x4 f32): lane L holds A[tile*16 + (L&15)][k + 2*(L>>4) + {0,1}]
// B fragment (4x16 f32): lane L holds Xcat[k + 2*(L>>4) + {0,1}][L&15]
//                        == XbT[L&15][k + 2*(L>>4) + {0,1}]  (contiguous pair)
// C/D (16x16 f32, 8 VGPRs): vgpr r -> row (tile*16 + r + 8*(L>>4)), col L&15.
// ---------------------------------------------------------------------------
__global__ __launch_bounds__(128) void gcn_aggregate(
        const float* __restrict__ A0, const float* __restrict__ A1,
        const float* __restrict__ A2, const float* __restrict__ xbt,
        float* __restrict__ Y) {
    int wave  = (int)((blockIdx.x * blockDim.x + threadIdx.x) >> 5); // 0..1535
    int lane  = threadIdx.x & 31;
    int layer = wave >> 9;     // 0..2  (512 tiles per layer)
    int tile  = wave & 511;

    const float* A = (layer == 0) ? A0 : ((layer == 1) ? A1 : A2);

    int half = lane >> 4;   // selects K pair {0,1} vs {2,3}
    int lo   = lane & 15;   // A row within tile / B column

    const float* arow = A   + (size_t)(tile * 16 + lo) * N_NODES + 2 * half;
    const float* brow = xbt + (size_t)lo * N_NODES + 2 * half;

    v8f c0 = {};
    v8f c1 = {};

#pragma unroll 4
    for (int k = 0; k < N_NODES; k += 8) {
        v2f a0 = *(const v2f*)(arow + k);
        v2f b0 = *(const v2f*)(brow + k);
        v2f a1v = *(const v2f*)(arow + k + 4);
        v2f b1v = *(const v2f*)(brow + k + 4);
        // 8 args: (neg_a, A, neg_b, B, c_mod, C, reuse_a, reuse_b)
        c0 = __builtin_amdgcn_wmma_f32_16x16x4_f32(false, a0, false, b0,
                                                   (short)0, c0, false, false);
        c1 = __builtin_amdgcn_wmma_f32_16x16x4_f32(false, a1v, false, b1v,
                                                   (short)0, c1, false, false);
    }
    v8f c = c0 + c1;

    float* yl = Y + (size_t)layer * N_NODES * NCOLS;
#pragma unroll
    for (int r = 0; r < 8; ++r) {
        yl[(size_t)(tile * 16 + r + 8 * half) * NCOLS + lo] = c[r];
    }
}

// ---------------------------------------------------------------------------
// Per-(node, batch) elementwise tail:
//   h_l = tanh(Y_l @ Wg_l + X @ Ws_l + b_l)
//   h   = tanh(h1 + h2 + h3)
//   Xn  = X + tanh(h @ Wout + bout)
// Writes Xn into next Xcat / XbT buffers and into the output slice.
// ---------------------------------------------------------------------------
__device__ __forceinline__ void gcs_layer(const float* y, const float* x,
                                          const float* __restrict__ Wg,
                                          const float* __restrict__ Ws,
                                          const float* __restrict__ bb,
                                          float* h) {
#pragma unroll
    for (int fo = 0; fo < FDIM; ++fo) {
        float u = bb[fo];
#pragma unroll
        for (int fi = 0; fi < FDIM; ++fi) {
            u += y[fi] * Wg[fi * FDIM + fo] + x[fi] * Ws[fi * FDIM + fo];
        }
        h[fo] = tanhf(u);
    }
}

__global__ __launch_bounds__(256) void gcn_update(
        const float* __restrict__ xcat, const float* __restrict__ Y,
        const float* __restrict__ Wg1, const float* __restrict__ Ws1,
        const float* __restrict__ b1,
        const float* __restrict__ Wg2, const float* __restrict__ Ws2,
        const float* __restrict__ b2,
        const float* __restrict__ Wg3, const float* __restrict__ Ws3,
        const float* __restrict__ b3,
        const float* __restrict__ Wout, const float* __restrict__ bout,
        float* __restrict__ xcat_next, float* __restrict__ xbt_next,
        float* __restrict__ out, int step, int pred_steps) {
    int t = blockIdx.x * blockDim.x + threadIdx.x;
    if (t >= N_NODES * BATCH) return;
    int b = t & (BATCH - 1);
    int n = t >> 2;

    float x[FDIM], y1[FDIM], y2[FDIM], y3[FDIM];
    size_t row = (size_t)n * NCOLS + b * FDIM;
#pragma unroll
    for (int f = 0; f < FDIM; ++f) {
        x[f]  = xcat[row + f];
        y1[f] = Y[row + f];
        y2[f] = Y[(size_t)N_NODES * NCOLS + row + f];
        y3[f] = Y[2 * (size_t)N_NODES * NCOLS + row + f];
    }

    float h1[FDIM], h2[FDIM], h3[FDIM], h[FDIM], xn[FDIM];
    gcs_layer(y1, x, Wg1, Ws1, b1, h1);
    gcs_layer(y2, x, Wg2, Ws2, b2, h2);
    gcs_layer(y3, x, Wg3, Ws3, b3, h3);
#pragma unroll
    for (int f = 0; f < FDIM; ++f) h[f] = tanhf(h1[f] + h2[f] + h3[f]);

#pragma unroll
    for (int fo = 0; fo < FDIM; ++fo) {
        float u = bout[fo];
#pragma unroll
        for (int fi = 0; fi < FDIM; ++fi) u += h[fi] * Wout[fi * FDIM + fo];
        xn[fo] = x[fo] + tanhf(u);
    }

#pragma unroll
    for (int f = 0; f < FDIM; ++f) {
        int c = b * FDIM + f;
        xcat_next[row + f] = xn[f];
        xbt_next[(size_t)c * N_NODES + n] = xn[f];
        // out[b][step][n][f]
        out[(((size_t)b * pred_steps + step) * N_NODES + n) * FDIM + f] = xn[f];
    }
}

// ---------------------------------------------------------------------------
// Host launcher. Inputs (setup_inputs order):
// 0 time_segs, 1 a1, 2 a2, 3 a3, 4 Wg1, 5 Ws1, 6 b1, 7 Wg2, 8 Ws2, 9 b2,
// 10 Wg3, 11 Ws3, 12 b3, 13 Wout, 14 bout, 15 pred_steps
// ---------------------------------------------------------------------------
extern "C" void kernel_launch(void* const* d_in, const int* in_sizes, int n_in,
                              void* d_out, int out_size, void* d_ws, size_t ws_size,
                              hipStream_t stream) {
    const float* ts   = (const float*)d_in[0];
    const float* a1   = (const float*)d_in[1];
    const float* a2   = (const float*)d_in[2];
    const float* a3   = (const float*)d_in[3];
    const float* Wg1  = (const float*)d_in[4];
    const float* Ws1  = (const float*)d_in[5];
    const float* b1   = (const float*)d_in[6];
    const float* Wg2  = (const float*)d_in[7];
    const float* Ws2  = (const float*)d_in[8];
    const float* b2   = (const float*)d_in[9];
    const float* Wg3  = (const float*)d_in[10];
    const float* Ws3  = (const float*)d_in[11];
    const float* b3   = (const float*)d_in[12];
    const float* Wout = (const float*)d_in[13];
    const float* bout = (const float*)d_in[14];

    // pred_steps is host-derivable: out = (B, pred_steps, N, F)
    int pred_steps = out_size / (BATCH * N_NODES * FDIM);
    if (pred_steps < 1) pred_steps = 1;

    const size_t XWORDS = (size_t)N_NODES * NCOLS; // 128K floats = 512 KB
    float* ws = (float*)d_ws;
    float* xcat[2] = { ws,              ws + XWORDS };
    float* xbt[2]  = { ws + 2 * XWORDS, ws + 3 * XWORDS };
    float* Y       = ws + 4 * XWORDS;              // 3 * N * 16 floats

    pack_x0<<<(N_NODES + 255) / 256, 256, 0, stream>>>(ts, xcat[0], xbt[0]);

    int cur = 0;
    for (int s = 0; s < pred_steps; ++s) {
        // 3 layers * 512 row-tiles = 1536 waves; 4 waves (128 thr) per block
        gcn_aggregate<<<384, 128, 0, stream>>>(a1, a2, a3, xbt[cur], Y);
        gcn_update<<<(N_NODES * BATCH + 255) / 256, 256, 0, stream>>>(
            xcat[cur], Y, Wg1, Ws1, b1, Wg2, Ws2, b2, Wg3, Ws3, b3,
            Wout, bout, xcat[1 - cur], xbt[1 - cur],
            (float*)d_out, s, pred_steps);
        cur = 1 - cur;
    }
}